// Sinkhorn_56762287784050
// MI455X (gfx1250) — compile-verified
//
#include <hip/hip_runtime.h>

#define SINKHORN_ITERS 20
#define N 64
#define STRIDE 66   // 64 + 2 pad: keeps 8B alignment, kills LDS bank conflicts

// ---- CDNA5 async global->LDS copy (8 bytes per lane) ------------------------
// VDST = 32-bit LDS byte address (low 32 bits of the generic shared pointer),
// VADDR = 64-bit global address. Tracked by ASYNCcnt.
__device__ __forceinline__ void async_copy_b64(const float* gsrc, float* ldst) {
    unsigned lds_addr = (unsigned)(unsigned long long)ldst;  // LDS byte offset
    asm volatile("global_load_async_to_lds_b64 %0, %1, off"
                 :: "v"(lds_addr), "v"(gsrc)
                 : "memory");
}

__device__ __forceinline__ void wait_async0() {
    asm volatile("s_wait_asynccnt 0" ::: "memory");
}

// One 64-thread block (2 wave32) per 64x64 matrix. Thread t owns column t.
__global__ __launch_bounds__(64) void sinkhorn_kernel(const float* __restrict__ x,
                                                      float* __restrict__ out) {
    __shared__ float buf[N * STRIDE];   // padded matrix tile (16,896 B)
    __shared__ float rs[N];             // per-row 1/rowsum broadcast slots

    const int t = threadIdx.x;                    // column owner, 0..63
    const float* xm = x   + (size_t)blockIdx.x * (N * N);
    float*       om = out + (size_t)blockIdx.x * (N * N);

    // ---- Async DMA: global matrix -> padded LDS (row r at byte r*STRIDE*4) ----
    // Step k copies 128 contiguous floats (2 rows): lane t moves 8 B.
    {
        const float* g = xm + 2 * t;
        float*       l = &buf[(t >> 5) * STRIDE + ((2 * t) & 63)];
        #pragma unroll
        for (int k = 0; k < 32; ++k) {
            async_copy_b64(g, l);
            g += 128;            // 512 B of global per step
            l += 2 * STRIDE;     // 2 padded rows of LDS per step
        }
    }
    wait_async0();
    __syncthreads();

    // ---- Load column t into registers; enter exp-space once. ----
    // Conflict-free: lanes read consecutive dwords of a row.
    float ev[N];
    #pragma unroll
    for (int i = 0; i < N; ++i) ev[i] = __expf(buf[i * STRIDE + t]);

    // ---- 20 Sinkhorn iterations, fully on-chip ----
    for (int it = 0; it < SINKHORN_ITERS; ++it) {
        // Column normalize (thread-local): ev /= sum(ev)
        float s0 = 0.f, s1 = 0.f, s2 = 0.f, s3 = 0.f;
        #pragma unroll
        for (int i = 0; i < N; i += 4) {
            s0 += ev[i]; s1 += ev[i + 1]; s2 += ev[i + 2]; s3 += ev[i + 3];
        }
        const float rcs = __builtin_amdgcn_rcpf((s0 + s1) + (s2 + s3));

        // Scale and scatter transposed: buf[row i][col t]. Lanes differ by
        // 1 dword per store -> conflict-free.
        #pragma unroll
        for (int i = 0; i < N; ++i) {
            ev[i] *= rcs;
            buf[i * STRIDE + t] = ev[i];
        }
        __syncthreads();

        // Row sum: thread t sums row t (contiguous, b64, conflict-free:
        // bank = (66t + 2j) mod 64 = 2t+2j distinct across a wave).
        const float2* row = (const float2*)&buf[t * STRIDE];
        float r0 = 0.f, r1 = 0.f;
        #pragma unroll
        for (int j = 0; j < N / 2; j += 2) {
            float2 a = row[j], b = row[j + 1];
            r0 += a.x + a.y;
            r1 += b.x + b.y;
        }
        rs[t] = __builtin_amdgcn_rcpf(r0 + r1);
        __syncthreads();

        // Row normalize: ev[i] *= 1/rowsum[i] (LDS broadcast reads, b128).
        #pragma unroll
        for (int i = 0; i < N; i += 4) {
            float4 rr = *(const float4*)&rs[i];
            ev[i]     *= rr.x;
            ev[i + 1] *= rr.y;
            ev[i + 2] *= rr.z;
            ev[i + 3] *= rr.w;
        }
    }

    // ---- Output: exp(log_softmax chain) == ev. Coalesced: each store covers
    // one full 256 B row across the 64 lanes. ----
    #pragma unroll
    for (int i = 0; i < N; ++i) om[i * N + t] = ev[i];
}

extern "C" void kernel_launch(void* const* d_in, const int* in_sizes, int n_in,
                              void* d_out, int out_size, void* d_ws, size_t ws_size,
                              hipStream_t stream) {
    (void)n_in; (void)out_size; (void)d_ws; (void)ws_size;
    const float* x = (const float*)d_in[0];
    float* out = (float*)d_out;
    const int n_mat = in_sizes[0] / (N * N);   // 8192
    sinkhorn_kernel<<<n_mat, N, 0, stream>>>(x, out);
}